// SparseLayer_36644660969843
// MI455X (gfx1250) — compile-verified
//
#include <hip/hip_runtime.h>

typedef __attribute__((ext_vector_type(2))) float v2f;
typedef __attribute__((ext_vector_type(4))) float v4f;
typedef __attribute__((ext_vector_type(8))) float v8f;

namespace {
constexpr int kD     = 128;   // D_IN == D_OUT
constexpr int kNeigh = 32;
constexpr int kRows  = 16;    // nodes per block
constexpr int kLda   = 132;   // padded LDS row stride (floats); 132*4=528B is 16B aligned
}

__global__ __launch_bounds__(256)
void sage_mean_wmma(const float* __restrict__ self_vecs,
                    const float* __restrict__ neigh_vecs,
                    const float* __restrict__ weight,   // [kD][kD] row-major
                    const float* __restrict__ bias,     // [kD]
                    float* __restrict__ out,            // [n][kD]
                    int n)
{
    __shared__ float sA[kRows * kLda];   // reduced & scaled A tile: 16 x 128 fp32

    const int tid = threadIdx.x;
    const int r0  = blockIdx.x * kRows;
    const float inv = 1.0f / (float)(kNeigh + 1);

    // ---------------- Phase 1: fused (self + sum of 32 neighbors) / 33 ----------------
    // 512 work items = 16 rows x 32 float4-columns; 256 threads -> 2 items each.
    #pragma unroll
    for (int p = 0; p < 2; ++p) {
        const int item = tid + p * 256;
        const int row  = item >> 5;          // 0..15
        const int c4   = item & 31;          // float4 column index (0..31)
        long long grow = (long long)r0 + row;
        if (grow >= n) grow = n - 1;         // clamp (redundant work; stores are exact-tiled)

        v4f acc = *(const v4f*)(self_vecs + grow * kD + c4 * 4);
        const v4f* np = (const v4f*)(neigh_vecs + grow * (long long)kNeigh * kD) + c4;
        #pragma unroll 8
        for (int j = 0; j < kNeigh; ++j) {
            // 819 MB one-shot stream: non-temporal so it doesn't evict W / L2 working set
            v4f v = __builtin_nontemporal_load(np + j * (kD / 4));
            acc += v;
        }
        acc *= inv;
        *(v4f*)&sA[row * kLda + c4 * 4] = acc;   // 16B-aligned -> ds_store_b128
    }

    __syncthreads();

    // ---------------- Phase 2: [16 x 128] @ [128 x 128] via V_WMMA_F32_16X16X4_F32 ----
    // 8 waves; wave w computes output columns [16w, 16w+16). K swept in chunks of 4.
    const int wave = tid >> 5;
    const int lane = tid & 31;
    const int l16  = lane & 15;
    const int koff = (lane & 16) ? 2 : 0;    // lanes 0-15 carry K=0,1; lanes 16-31 carry K=2,3
    const int ncol = wave * 16 + l16;

    v8f c = {};  // f32 accumulator, 8 VGPRs

    const float* aP = &sA[l16 * kLda + koff];        // A: row = lane%16 (padded stride: no bank conflict)
    const float* bP = weight + (long long)koff * kD + ncol; // B: W[koff + k][ncol], cache-resident

    #pragma unroll
    for (int k = 0; k < kD; k += 4) {
        v2f a = *(const v2f*)(aP + k);               // ds_load_b64 (8B aligned)
        v2f b;
        b.x = bP[(k + 0) * kD];
        b.y = bP[(k + 1) * kD];
        c = __builtin_amdgcn_wmma_f32_16x16x4_f32(
                /*neg_a=*/false, a, /*neg_b=*/false, b,
                /*c_mod=*/(short)0, c, /*reuse_a=*/false, /*reuse_b=*/false);
    }

    // ---------------- Epilogue: bias + store (C layout: VGPR v -> M=v(+8), N=lane%16) --
    const float bv = bias[ncol];
    const int mbase = (lane & 16) ? 8 : 0;
    #pragma unroll
    for (int v = 0; v < 8; ++v) {
        const long long grow = (long long)r0 + mbase + v;
        if (grow < n) out[grow * kD + ncol] = c[v] + bv;
    }
}

extern "C" void kernel_launch(void* const* d_in, const int* in_sizes, int n_in,
                              void* d_out, int out_size, void* d_ws, size_t ws_size,
                              hipStream_t stream) {
    // setup_inputs order: self_vecs, neigh_vecs, neigh_num(scalar), weight, bias
    const float* self_vecs  = (const float*)d_in[0];
    const float* neigh_vecs = (const float*)d_in[1];
    const float* weight     = (const float*)d_in[3];
    const float* bias       = (const float*)d_in[4];
    float* out = (float*)d_out;

    const int n = in_sizes[0] / kD;                // N = 50000
    const int grid = (n + kRows - 1) / kRows;      // 3125 blocks
    hipLaunchKernelGGL(sage_mean_wmma, dim3(grid), dim3(256), 0, stream,
                       self_vecs, neigh_vecs, weight, bias, out, n);
}